// ForuierChannelMixer3_28338194219514
// MI455X (gfx1250) — compile-verified
//
#include <hip/hip_runtime.h>

// ---------------------------------------------------------------------------
// FourierChannelMixer (rfft32 -> complex sine gate -> irfft32) for MI455X.
//
// Both DFTs are real 32x32 GEMMs on V_WMMA_F32_16X16X4_F32 (wave32, full f32).
// Spectrum packing (exact, since Im(bin0)=Im(bin16)=0 for real input and
// irfft ignores Im of bins 0/16):
//   col 0..16  = Re(bin 0..16),  col 17..31 = Im(bin 1..15)
//
// HBM-bound: 128 MiB traffic => ~5.5us floor @ 23.3 TB/s. WMMA carries the
// ~2.3 GFLOP of DFT matmul; VALU only does the complex-sine transcendentals.
// DFT matrices are synthesized per-lane with hardware v_sin/v_cos (angles are
// multiples of pi/16, index-reduced mod 32) so setup is ~150 VALU instrs.
// ---------------------------------------------------------------------------

typedef float v2f __attribute__((ext_vector_type(2)));
typedef float v4f __attribute__((ext_vector_type(4)));
typedef float v8f __attribute__((ext_vector_type(8)));

#define PI_16      0.19634954084936207740f   // pi/16 = (2*pi)/32
#define INV_SQRT32 0.17677669529663688110f
#define LROW 36   // LDS row stride (dwords): 144B rows -> 16B aligned, spreads banks

__device__ __forceinline__ float tcos(int idx) {   // cos(2*pi*idx/32)
  return __cosf((float)(idx & 31) * PI_16);
}
__device__ __forceinline__ float tsin(int idx) {   // sin(2*pi*idx/32)
  return __sinf((float)(idx & 31) * PI_16);
}

__device__ __forceinline__ float b2_elem(int k, int col) {
  // Inverse-DFT matrix row k, output channel col.
  // out[n] = inv*[Wr0 + (-1)^n Wr16 + 2*sum_f (Wrf cos - Wif sin)]
  if (k <= 16) {
    float coef = (k == 0 || k == 16) ? 1.0f : 2.0f;
    return coef * tcos(k * col) * INV_SQRT32;
  } else {
    return -2.0f * tsin((k - 16) * col) * INV_SQRT32;
  }
}

__global__ __launch_bounds__(256) void fcm3_wmma_kernel(
    const float* __restrict__ x,
    const float* __restrict__ c1r_p, const float* __restrict__ c1i_p,
    const float* __restrict__ c2r_p, const float* __restrict__ c2i_p,
    float* __restrict__ out, int ntiles)
{
  __shared__ float lds[8][16 * LROW];   // per-wave staging (8 waves/block)

  const int lane = threadIdx.x & 31;
  const int wave = threadIdx.x >> 5;
  const int n    = lane & 15;   // column-in-tile == frequency bin for this lane
  const int h    = lane >> 4;   // half-wave select (K sub-pair / row group)
  float* myLds = &lds[wave][0];

  // ---- build constant DFT fragments once per lane (hw trig) ----------------
  // Assumed 16x4 f32 A/B layout: lanes 0-15 v0={K=0},v1={K=1}; lanes 16-31
  // v0={K=2},v1={K=3} (B mirrors A with N striped across lanes).
  v2f b1t0[8], b1t1[8], b2t0[8], b2t1[8];
  #pragma unroll
  for (int c = 0; c < 8; ++c) {
    const int k0 = c * 4 + 2 * h;
    #pragma unroll
    for (int q = 0; q < 2; ++q) {
      const int k = k0 + q;
      // forward: col n -> Re(bin n); col 16+n -> (n==0 ? Re bin16 : Im bin n)
      b1t0[c][q] = tcos(k * n) * INV_SQRT32;
      b1t1[c][q] = (n == 0) ? tcos(k * 16) * INV_SQRT32
                            : -tsin(k * n) * INV_SQRT32;
      // inverse: K index k of packed W -> output channels n and 16+n
      b2t0[c][q] = b2_elem(k, n);
      b2t1[c][q] = b2_elem(k, 16 + n);
    }
  }

  // gate parameters: this lane's bin is n; lanes with n==0 also own bin 16
  const float c1r = c1r_p[n],  c1i = c1i_p[n];
  const float c2r = c2r_p[n],  c2i = c2i_p[n];
  const float c1r16 = c1r_p[16], c1i16 = c1i_p[16];
  const float c2r16 = c2r_p[16], c2i16 = c2i_p[16];

  const int waveGlobal = (int)((blockIdx.x * blockDim.x + threadIdx.x) >> 5);
  const int nWaves     = (int)(gridDim.x * (blockDim.x >> 5));

  // ---- software-pipelined grid-stride loop over 16-row tiles ---------------
  int tile = waveGlobal;
  v4f cur[4] = {};
  if (tile < ntiles) {
    const float* xt = x + (size_t)tile * 512;
    #pragma unroll
    for (int it = 0; it < 4; ++it)
      cur[it] = *(const v4f*)(xt + (size_t)(lane + 32 * it) * 4);
  }

  for (; tile < ntiles; tile += nWaves) {
    // ---- stage current tile into LDS --------------------------------------
    #pragma unroll
    for (int it = 0; it < 4; ++it) {
      const int i4  = lane + 32 * it;           // float4 index within tile
      const int row = i4 >> 3, c4 = i4 & 7;
      *(v4f*)(myLds + row * LROW + c4 * 4) = cur[it];
    }

    // ---- prefetch next tile (overlaps with WMMA + transcendentals) ---------
    const int nextTile = tile + nWaves;
    v4f nxt[4] = {};
    if (nextTile < ntiles) {
      const float* xn = x + (size_t)nextTile * 512;
      #pragma unroll
      for (int it = 0; it < 4; ++it)
        nxt[it] = *(const v4f*)(xn + (size_t)(lane + 32 * it) * 4);
    }

    asm volatile("s_wait_dscnt 0" ::: "memory");

    // ---- A fragments: lane holds row n, K pair (4c+2h, 4c+2h+1) ------------
    v2f a[8];
    #pragma unroll
    for (int c = 0; c < 8; ++c)
      a[c] = *(const v2f*)(myLds + n * LROW + c * 4 + 2 * h);

    // ---- GEMM1: Y = X @ B1  (16 x v_wmma_f32_16x16x4_f32) ------------------
    v8f y0 = {}, y1 = {};
    #pragma unroll
    for (int c = 0; c < 8; ++c) {
      y0 = __builtin_amdgcn_wmma_f32_16x16x4_f32(false, a[c], false, b1t0[c],
                                                 (short)0, y0, false, false);
      y1 = __builtin_amdgcn_wmma_f32_16x16x4_f32(false, a[c], false, b1t1[c],
                                                 (short)0, y1, false, false);
    }

    // ---- complex sine gate: w = sin(y*cm1)*cm2, scatter W back to LDS ------
    // D layout: vgpr j holds row j+8h, lane's column n (tile0) / 16+n (tile1)
    #pragma unroll
    for (int j = 0; j < 8; ++j) {
      const float yr  = y0[j];
      const float t1v = y1[j];
      const float yi  = (n == 0) ? 0.0f : t1v;
      const float zr  = yr * c1r - yi * c1i;
      const float zi  = yr * c1i + yi * c1r;
      const float ex  = __expf(zi), emx = __expf(-zi);
      const float ch  = 0.5f * (ex + emx), sh = 0.5f * (ex - emx);
      const float sr  = __sinf(zr) * ch;
      const float si  = __cosf(zr) * sh;
      const float wr  = sr * c2r - si * c2i;
      float o1;
      if (n == 0) {               // this lane also carries Re(bin 16) = t1v
        const float z2r = t1v * c1r16, z2i = t1v * c1i16;
        const float e2 = __expf(z2i), em2 = __expf(-z2i);
        const float ch2 = 0.5f * (e2 + em2), sh2 = 0.5f * (e2 - em2);
        o1 = (__sinf(z2r) * ch2) * c2r16 - (__cosf(z2r) * sh2) * c2i16;
      } else {
        o1 = sr * c2i + si * c2r; // Im part
      }
      const int row = j + 8 * h;
      myLds[row * LROW + n]      = wr;   // packed W, col n
      myLds[row * LROW + 16 + n] = o1;   // packed W, col 16+n
    }
    asm volatile("s_wait_dscnt 0" ::: "memory");

    // ---- GEMM2: Out = W @ B2 ----------------------------------------------
    v2f w[8];
    #pragma unroll
    for (int c = 0; c < 8; ++c)
      w[c] = *(const v2f*)(myLds + n * LROW + c * 4 + 2 * h);

    v8f o0 = {}, o1v = {};
    #pragma unroll
    for (int c = 0; c < 8; ++c) {
      o0  = __builtin_amdgcn_wmma_f32_16x16x4_f32(false, w[c], false, b2t0[c],
                                                  (short)0, o0, false, false);
      o1v = __builtin_amdgcn_wmma_f32_16x16x4_f32(false, w[c], false, b2t1[c],
                                                  (short)0, o1v, false, false);
    }

    // ---- store: lanes 0-15 cover cols 0-15 of row j, lanes 16-31 row j+8 ---
    float* ot = out + (size_t)tile * 512;
    #pragma unroll
    for (int j = 0; j < 8; ++j) {
      const int row = j + 8 * h;
      ot[row * 32 + n]      = o0[j];
      ot[row * 32 + 16 + n] = o1v[j];
    }

    // rotate prefetch (same-wave DS is in-order => safe to re-stage next iter)
    #pragma unroll
    for (int it = 0; it < 4; ++it) cur[it] = nxt[it];
  }
}

extern "C" void kernel_launch(void* const* d_in, const int* in_sizes, int n_in,
                              void* d_out, int out_size, void* d_ws, size_t ws_size,
                              hipStream_t stream) {
  (void)in_sizes; (void)n_in; (void)d_ws; (void)ws_size;
  const float* x   = (const float*)d_in[0];
  const float* c1r = (const float*)d_in[1];
  const float* c1i = (const float*)d_in[2];
  const float* c2r = (const float*)d_in[3];
  const float* c2i = (const float*)d_in[4];
  float* out = (float*)d_out;

  const int ntiles = out_size / 512;          // 16 rows x 32 channels per tile
  int blocks = (ntiles + 7) / 8;              // 8 waves per 256-thread block
  if (blocks > 2048) blocks = 2048;           // ~2 tiles/wave: amortize + prefetch
  if (blocks < 1) blocks = 1;
  fcm3_wmma_kernel<<<blocks, 256, 0, stream>>>(x, c1r, c1i, c2r, c2i, out, ntiles);
}